// SandwichConvLin_798863917434
// MI455X (gfx1250) — compile-verified
//
#include <hip/hip_runtime.h>
#include <math.h>

// ---------------------------------------------------------------------------
// SandwichConvLin on gfx1250: everything GEMM-shaped runs on V_WMMA_F32_16X16X4_F32
// (fp32 matrix pipe; fp32 needed because the Cayley inverse amplifies low-precision
// error). One workgroup (8 x wave32) owns one frequency bin, fully LDS-resident.
// Async LDS engine (ASYNCcnt) hides the xfft tile fetch behind the Gauss-Jordan
// phase and streams the yfft tile out at kernel end.
// ---------------------------------------------------------------------------

typedef __attribute__((ext_vector_type(2))) float v2f;
typedef __attribute__((ext_vector_type(8))) float v8f;
typedef __attribute__((ext_vector_type(4))) int   v4i;

#define NN    64
#define NF    33
#define NFREQ (NN * NF)          // 2112
#define BATCH 32
#define CIN   64
#define COUT  64
#define CHN   128

// workspace layout (float offsets)
#define OFF_F    0                       // Fr[4096] then Fi[4096]
#define OFF_NORM 8192                    // 1 float (sum of |wfft|^2)
#define OFF_XR   8448
#define PLANE    (NFREQ * 64 * 32)       // 4,325,376 floats
#define OFF_XI   (OFF_XR + PLANE)
#define OFF_YR   (OFF_XI + PLANE)
#define OFF_YI   (OFF_YR + PLANE)

// ---- async LDS engine (gfx1250), guarded so absence degrades to sync path ----
#if defined(__AMDGCN__) && \
    __has_builtin(__builtin_amdgcn_global_load_async_to_lds_b128) && \
    __has_builtin(__builtin_amdgcn_s_wait_asynccnt)
#define HAVE_ASYNC_LOAD 1
#else
#define HAVE_ASYNC_LOAD 0
#endif
#if defined(__AMDGCN__) && \
    __has_builtin(__builtin_amdgcn_global_store_async_from_lds_b128)
#define HAVE_ASYNC_STORE 1
#else
#define HAVE_ASYNC_STORE 0
#endif

#define AS1 __attribute__((address_space(1)))
#define AS3 __attribute__((address_space(3)))

#if HAVE_ASYNC_LOAD
template <int IOFF>
__device__ __forceinline__ void async_g2l_b128(const void* g, const void* l) {
  // builtin signature (from clang diagnostic): (int4 AS1*, int4 AS3*, imm, imm)
  __builtin_amdgcn_global_load_async_to_lds_b128(
      (AS1 v4i*)(uintptr_t)g,
      (AS3 v4i*)(unsigned int)(uintptr_t)l, IOFF, 0);
}
#endif
#if HAVE_ASYNC_STORE
template <int IOFF>
__device__ __forceinline__ void async_l2g_b128(void* g, const void* l) {
  __builtin_amdgcn_global_store_async_from_lds_b128(
      (AS1 v4i*)(uintptr_t)g,
      (AS3 v4i*)(unsigned int)(uintptr_t)l, IOFF, 0);
}
#endif

__device__ __forceinline__ v8f wmma4(v2f a, v2f b, v8f c) {
  // D = A(16x4) * B(4x16) + C, fp32 in / fp32 out.
  return __builtin_amdgcn_wmma_f32_16x16x4_f32(
      false, a, false, b, (short)0, c, false, false);
}

// ---------------------------------------------------------------------------
// Generic (possibly complex) GEMM on LDS operands, tiled 16x16 per wave.
//   C(MxN) = scale * opA(A) * opB(B)
// TA/TB: operand is stored transposed (access [k][m] instead of [m][k]).
// CA/CB: operand has an imaginary plane. sA/sB = +-1 implement conjugation.
// Ci may be nullptr to drop the imaginary output.
// K must be a multiple of 4; M,N multiples of 16. Uniform control flow per
// wave so EXEC is all-ones at every WMMA (ISA requirement).
// ---------------------------------------------------------------------------
template <bool TA, bool TB, bool CA, bool CB>
__device__ void cgemm(int M, int N, int K,
                      const float* __restrict__ Ar, const float* __restrict__ Ai, int lda,
                      const float* __restrict__ Br, const float* __restrict__ Bi, int ldb,
                      float* __restrict__ Cr, float* __restrict__ Ci, int ldc,
                      float sA, float sB, float scale) {
  const int lane   = threadIdx.x & 31;
  const int wave   = threadIdx.x >> 5;
  const int nwaves = blockDim.x >> 5;
  const int mn  = lane & 15;            // row index for A frag, col index for B frag
  const int kh  = (lane >> 4) << 1;     // lanes 0-15 -> K 0,1 ; lanes 16-31 -> K 2,3
  const int tgn = N >> 4;
  const int ntile = (M >> 4) * tgn;

  for (int t = wave; t < ntile; t += nwaves) {
    const int bm = (t / tgn) << 4;
    const int bn = (t % tgn) << 4;
    v8f rr = {0.f,0.f,0.f,0.f,0.f,0.f,0.f,0.f};
    v8f ri = rr, ir = rr, ii = rr;
    for (int k0 = 0; k0 < K; k0 += 4) {
      const int ka = k0 + kh;
      v2f ar, ai, br, bi;
      ar.x = ar.y = ai.x = ai.y = 0.f;
      br.x = br.y = bi.x = bi.y = 0.f;
      if (TA) { ar.x = Ar[ ka   *lda + bm + mn]; ar.y = Ar[(ka+1)*lda + bm + mn]; }
      else    { ar.x = Ar[(bm+mn)*lda + ka];     ar.y = Ar[(bm+mn)*lda + ka+1]; }
      if (CA) {
        if (TA) { ai.x = Ai[ ka   *lda + bm + mn]; ai.y = Ai[(ka+1)*lda + bm + mn]; }
        else    { ai.x = Ai[(bm+mn)*lda + ka];     ai.y = Ai[(bm+mn)*lda + ka+1]; }
      }
      if (TB) { br.x = Br[(bn+mn)*ldb + ka];     br.y = Br[(bn+mn)*ldb + ka+1]; }
      else    { br.x = Br[ ka   *ldb + bn + mn]; br.y = Br[(ka+1)*ldb + bn + mn]; }
      if (CB) {
        if (TB) { bi.x = Bi[(bn+mn)*ldb + ka];     bi.y = Bi[(bn+mn)*ldb + ka+1]; }
        else    { bi.x = Bi[ ka   *ldb + bn + mn]; bi.y = Bi[(ka+1)*ldb + bn + mn]; }
      }
      rr = wmma4(ar, br, rr);
      if (CB)       ri = wmma4(ar, bi, ri);
      if (CA)       ir = wmma4(ai, br, ir);
      if (CA && CB) ii = wmma4(ai, bi, ii);
    }
    // combine with conj signs (f32 WMMA has no A/B NEG per ISA; fold here)
    const int mb = bm + ((lane >> 4) << 3);
#pragma unroll
    for (int r = 0; r < 8; ++r) {
      float cr = rr[r];
      float cim = 0.f;
      if (CA && CB) cr -= sA * sB * ii[r];
      if (CA)       cim += sA * ir[r];
      if (CB)       cim += sB * ri[r];
      Cr[(mb + r) * ldc + bn + mn] = scale * cr;
      if (Ci) Ci[(mb + r) * ldc + bn + mn] = scale * cim;
    }
  }
}

// ---------------------------------------------------------------------------
// DFT matrix F[u][a] = exp(-2*pi*i*u*a/64)
// ---------------------------------------------------------------------------
__global__ void __launch_bounds__(256) k_twiddle(float* ws) {
  int i = blockIdx.x * blockDim.x + threadIdx.x;
  if (i < NN * NN) {
    int u = i >> 6, a = i & 63;
    float ang = -2.0f * (float)M_PI * (float)((u * a) & 63) / 64.0f;
    ws[OFF_F + i]        = cosf(ang);
    ws[OFF_F + 4096 + i] = sinf(ang);
  }
}

// ---------------------------------------------------------------------------
// ||wfft||_F^2 : 3x3 kernel -> 9-term phase sum per spectral element.
// ---------------------------------------------------------------------------
__global__ void __launch_bounds__(256) k_norm(const float* __restrict__ w, float* ws) {
  int f = blockIdx.x;
  int u = f / NF, v = f % NF;
  const float th = 2.0f * (float)M_PI / 64.0f;
  float cu, su, cv, sv;
  __sincosf(th * (float)u, &su, &cu);
  __sincosf(th * (float)v, &sv, &cv);
  // e^{-i*theta*u*k1}, k1 = 0,1,2
  float pur[3] = {1.f, cu, cu*cu - su*su};
  float pui[3] = {0.f, -su, -2.f*su*cu};
  float pvr[3] = {1.f, cv, cv*cv - sv*sv};
  float pvi[3] = {0.f, -sv, -2.f*sv*cv};
  float acc = 0.f;
  for (int e = 0; e < 32; ++e) {
    int idx = threadIdx.x + (e << 8);        // 8192 = cout*chn
    int co = idx >> 7, ch = idx & 127;
    const float* wp = w + (co * CHN + ch) * 9;
    float ar = 0.f, ai = 0.f;
#pragma unroll
    for (int k1 = 0; k1 < 3; ++k1)
#pragma unroll
      for (int k2 = 0; k2 < 3; ++k2) {
        float pr = pur[k1]*pvr[k2] - pui[k1]*pvi[k2];
        float pi = pur[k1]*pvi[k2] + pui[k1]*pvr[k2];
        float wv = wp[k1*3 + k2];
        ar += wv * pr; ai += wv * pi;
      }
    acc += ar*ar + ai*ai;
  }
  __shared__ float red[256];
  red[threadIdx.x] = acc;
  __syncthreads();
  for (int s = 128; s > 0; s >>= 1) {
    if (threadIdx.x < s) red[threadIdx.x] += red[threadIdx.x + s];
    __syncthreads();
  }
  if (threadIdx.x == 0) atomicAdd(ws + OFF_NORM, red[0]);
}

// ---------------------------------------------------------------------------
// rfft2(x) as Y = F * x * F^T (WMMA GEMMs); store cols 0..32 into (f,cin,B).
// ---------------------------------------------------------------------------
__global__ void __launch_bounds__(256) k_fft(const float* __restrict__ x, float* ws) {
  extern __shared__ float sm[];
  float* xs = sm;               // 4096
  float* Fr = xs + 4096;
  float* Fi = Fr + 4096;
  float* Tr = Fi + 4096;
  float* Ti = Tr + 4096;
  float* Yr = Ti + 4096;
  float* Yi = Yr + 4096;        // total 7*4096 floats = 112KB
  int b = blockIdx.x >> 6, c = blockIdx.x & 63;
  const float* xg = x + (((size_t)(b * CIN + c)) << 12);
  for (int e = threadIdx.x; e < 4096; e += 256) {
    xs[e] = xg[e];
    Fr[e] = ws[OFF_F + e];
    Fi[e] = ws[OFF_F + 4096 + e];
  }
  __syncthreads();
  // T = F * x  (complex * real)
  cgemm<false,false,true,false>(64,64,64, Fr,Fi,64, xs,nullptr,64, Tr,Ti,64, 1.f,0.f,1.f);
  __syncthreads();
  // Y = T * F^T (complex * complex)
  cgemm<false,true,true,true>(64,64,64, Tr,Ti,64, Fr,Fi,64, Yr,Yi,64, 1.f,1.f,1.f);
  __syncthreads();
  for (int e = threadIdx.x; e < NFREQ; e += 256) {
    int u = e / 33, v = e - u * 33;
    size_t g = (size_t)e * 2048 + c * 32 + b;
    ws[OFF_XR + g] = Yr[u*64 + v];
    ws[OFF_XI + g] = Yi[u*64 + v];
  }
}

// ---------------------------------------------------------------------------
// Per-frequency Cayley + application. One workgroup per frequency bin.
//   Wt[ch][co] = (alpha/||wfft||) * sum_{k1,k2} w[co][ch][k1][k2] e^{i th(u(k1-1)+v(k2-1))}
//   U = Wt[:64], V = Wt[64:], A = U - U^H + V^H V, N = inv(I+A), P = N(I-A)
//   yfft = -4 * conj(P) * (N^T * (V^T * xfft))
// The xfft tile is fetched by the async-LDS engine at kernel entry and only
// waited on after the inversion: the load is fully hidden behind ~9 MFLOP of
// WMMA/VALU work, costs no VGPRs, and lands directly in LDS.
// ---------------------------------------------------------------------------
__global__ void __launch_bounds__(256) k_cayley(const float* __restrict__ w,
                                                const float* __restrict__ alpha,
                                                float* ws) {
  extern __shared__ float sm[];
  float* Ur = sm;              float* Ui = Ur + 4096;
  float* Vr = Ui + 4096;       float* Vi = Vr + 4096;
  float* Mr = Vi + 4096;       float* Mi = Mr + 4096;   // A, then M = I+A (GJ in place)
  float* Nr = Mi + 4096;       float* Ni = Nr + 4096;   // identity -> inverse
  float* Gr = Ni + 4096;       float* Gi = Gr + 4096;   // I - A ; later yfft tile
  float* Xr = Gi + 4096;       float* Xi = Xr + 2048;   // x tile ; later h tile
  float* Sr = Xi + 2048;       float* Si = Sr + 2048;   // s = V^T x
  __shared__ float facR[64], facI[64];

  const int f = blockIdx.x;
  const int u = f / NF, v = f % NF;
  const int tid = threadIdx.x;

#if HAVE_ASYNC_LOAD
  // kick off the xfft tile copy (2 x 8KB) into LDS; consumed ~after the GJ phase
  {
    const float* gR = ws + OFF_XR + (size_t)f * 2048 + tid * 4;
    const float* gI = ws + OFF_XI + (size_t)f * 2048 + tid * 4;
    async_g2l_b128<0>   (gR, Xr + tid * 4);
    async_g2l_b128<4096>(gR, Xr + tid * 4);   // +1024 floats on both sides
    async_g2l_b128<0>   (gI, Xi + tid * 4);
    async_g2l_b128<4096>(gI, Xi + tid * 4);
  }
#endif

  __builtin_prefetch(w + tid * 9, 0, 0);   // global_prefetch_b8: weights are L2-hot

  const float sc = alpha[0] * rsqrtf(ws[OFF_NORM]);
  const float th = 2.0f * (float)M_PI / 64.0f;
  float cu, su, cv, sv;
  __sincosf(th * (float)u, &su, &cu);
  __sincosf(th * (float)v, &sv, &cv);
  // e^{i*theta*u*d}, d = -1, 0, +1
  float pur[3] = { cu, 1.f, cu };
  float pui[3] = { -su, 0.f, su };
  float pvr[3] = { cv, 1.f, cv };
  float pvi[3] = { -sv, 0.f, sv };

  // ---- build U (ch<64) and V (ch>=64) ----
  for (int e = 0; e < 32; ++e) {
    int idx = tid + (e << 8);              // 8192 elements
    int r = idx & 127, c = idx >> 7;       // r = ch, c = co
    const float* wp = w + (c * CHN + r) * 9;
    float ar = 0.f, ai = 0.f;
#pragma unroll
    for (int k1 = 0; k1 < 3; ++k1)
#pragma unroll
      for (int k2 = 0; k2 < 3; ++k2) {
        float pr = pur[k1]*pvr[k2] - pui[k1]*pvi[k2];
        float pi = pur[k1]*pvi[k2] + pui[k1]*pvr[k2];
        float wv = wp[k1*3 + k2];
        ar += wv * pr; ai += wv * pi;
      }
    ar *= sc; ai *= sc;
    if (r < 64) { Ur[r*64 + c] = ar; Ui[r*64 + c] = ai; }
    else        { Vr[(r-64)*64 + c] = ar; Vi[(r-64)*64 + c] = ai; }
  }
  __syncthreads();

  // ---- A = V^H V (WMMA) ----
  cgemm<true,false,true,true>(64,64,64, Vr,Vi,64, Vr,Vi,64, Mr,Mi,64, -1.f,1.f,1.f);
  __syncthreads();

  // ---- A += U - U^H ; G = I - A ; M = I + A ; N = I ----
  for (int e = 0; e < 16; ++e) {
    int idx = tid + (e << 8);              // 4096
    int i = idx >> 6, j = idx & 63;
    float ar = Mr[idx] + Ur[i*64 + j] - Ur[j*64 + i];
    float ai = Mi[idx] + Ui[i*64 + j] + Ui[j*64 + i];
    float d = (i == j) ? 1.f : 0.f;
    Gr[idx] = d - ar;  Gi[idx] = -ai;
    Mr[idx] = d + ar;  Mi[idx] = ai;
    Nr[idx] = d;       Ni[idx] = 0.f;
  }
  __syncthreads();

  // ---- Gauss-Jordan: [M | N] -> [I | inv(M)]  (complex, no pivoting) ----
  for (int k = 0; k < 64; ++k) {
    float pr = Mr[k*64 + k], pi = Mi[k*64 + k];
    float inv = 1.f / (pr*pr + pi*pi);
    float qr = pr * inv, qi = -pi * inv;
    if (tid >= 128 && tid < 192) {         // stash column k (untouched by row-scale)
      int i = tid - 128;
      facR[i] = Mr[i*64 + k]; facI[i] = Mi[i*64 + k];
    }
    __syncthreads();
    if (tid < 128) {                       // scale pivot row of [M|N]
      int j = tid & 63;
      float* Rr = (tid < 64) ? Mr : Nr;
      float* Ri = (tid < 64) ? Mi : Ni;
      float xr = Rr[k*64 + j], xi2 = Ri[k*64 + j];
      Rr[k*64 + j] = xr*qr - xi2*qi;
      Ri[k*64 + j] = xr*qi + xi2*qr;
    }
    __syncthreads();
    for (int e = 0; e < 32; ++e) {         // eliminate
      int idx = tid + (e << 8);            // 8192 = 64 rows x 128 cols
      int i = idx >> 7, j = idx & 127;
      if (i == k) continue;
      float cr = facR[i], ci = facI[i];
      float* Rr = (j < 64) ? Mr : Nr;
      float* Ri = (j < 64) ? Mi : Ni;
      int jj = j & 63;
      float rkr = Rr[k*64 + jj], rki = Ri[k*64 + jj];
      Rr[i*64 + jj] -= cr*rkr - ci*rki;
      Ri[i*64 + jj] -= cr*rki + ci*rkr;
    }
    __syncthreads();
  }

  // ---- P = N * (I - A) -> reuse U region ----
  cgemm<false,false,true,true>(64,64,64, Nr,Ni,64, Gr,Gi,64, Ur,Ui,64, 1.f,1.f,1.f);
  __syncthreads();

  // ---- xfft tile ready? (async engine) ----
#if HAVE_ASYNC_LOAD
  __builtin_amdgcn_s_wait_asynccnt(0);
  __syncthreads();
#else
  for (int e = tid; e < 2048; e += 256) {
    size_t g = (size_t)f * 2048 + e;
    Xr[e] = ws[OFF_XR + g];
    Xi[e] = ws[OFF_XI + g];
  }
  __syncthreads();
#endif

  // ---- s = V^T x ----
  cgemm<true,false,true,true>(64,32,64, Vr,Vi,64, Xr,Xi,32, Sr,Si,32, 1.f,1.f,1.f);
  __syncthreads();
  // ---- h = N^T s -> reuse X region ----
  cgemm<true,false,true,true>(64,32,64, Nr,Ni,64, Sr,Si,32, Xr,Xi,32, 1.f,1.f,1.f);
  __syncthreads();
  // ---- yfft = -4 * conj(P) * h -> reuse G region ----
  cgemm<false,false,true,true>(64,32,64, Ur,Ui,64, Xr,Xi,32, Gr,Gi,32, -1.f,1.f,-4.f);
  __syncthreads();

#if HAVE_ASYNC_STORE
  // stream yfft tile straight from LDS; S_ENDPGM's implicit wait-idle covers ASYNCcnt
  {
    float* gR = ws + OFF_YR + (size_t)f * 2048 + tid * 4;
    float* gI = ws + OFF_YI + (size_t)f * 2048 + tid * 4;
    async_l2g_b128<0>   (gR, Gr + tid * 4);
    async_l2g_b128<4096>(gR, Gr + tid * 4);
    async_l2g_b128<0>   (gI, Gi + tid * 4);
    async_l2g_b128<4096>(gI, Gi + tid * 4);
  }
#else
  for (int e = tid; e < 2048; e += 256) {
    size_t g = (size_t)f * 2048 + e;
    ws[OFF_YR + g] = Gr[e];
    ws[OFF_YI + g] = Gi[e];
  }
#endif
}

// ---------------------------------------------------------------------------
// irfft2 as y = real( C * Yfull * C^T ), C = conj(F)/64; Hermitian rebuild of
// the missing 31 columns; + bias.
// ---------------------------------------------------------------------------
__global__ void __launch_bounds__(256) k_out(const float* __restrict__ bias,
                                             const float* __restrict__ ws,
                                             float* __restrict__ out) {
  extern __shared__ float sm[];
  float* Fr = sm;          float* Fi = Fr + 4096;
  float* Yr = Fi + 4096;   float* Yi = Yr + 4096;
  float* Tr = Yi + 4096;   float* Ti = Tr + 4096;
  float* Or = Ti + 4096;   // 4096 ; total 7*4096 floats
  int b = blockIdx.x >> 6, co = blockIdx.x & 63;
  for (int e = threadIdx.x; e < 4096; e += 256) {
    Fr[e] = ws[OFF_F + e];
    Fi[e] = ws[OFF_F + 4096 + e];
  }
  for (int e = threadIdx.x; e < NFREQ; e += 256) {
    int u = e / 33, v = e - u * 33;
    size_t g = (size_t)e * 2048 + co * 32 + b;
    Yr[u*64 + v] = ws[OFF_YR + g];
    Yi[u*64 + v] = ws[OFF_YI + g];
  }
  __syncthreads();
  for (int e = threadIdx.x; e < 2048; e += 256) {   // Hermitian fill v = 33..63
    int u2 = e >> 5, vv = e & 31;
    if (vv < 31) {
      int v2 = 33 + vv;
      int us = (64 - u2) & 63, vs = 64 - v2;
      Yr[u2*64 + v2] =  Yr[us*64 + vs];
      Yi[u2*64 + v2] = -Yi[us*64 + vs];
    }
  }
  __syncthreads();
  // T = conj(F)/64 * Y
  cgemm<false,false,true,true>(64,64,64, Fr,Fi,64, Yr,Yi,64, Tr,Ti,64, -1.f,1.f, 1.f/64.f);
  __syncthreads();
  // y = real( T * conj(F)^T/64 )
  cgemm<false,true,true,true>(64,64,64, Tr,Ti,64, Fr,Fi,64, Or,nullptr,64, 1.f,-1.f, 1.f/64.f);
  __syncthreads();
  float bv = bias[co];
  float* og = out + (((size_t)(b * COUT + co)) << 12);
  for (int e = threadIdx.x; e < 4096; e += 256) og[e] = Or[e] + bv;
}

// ---------------------------------------------------------------------------
extern "C" void kernel_launch(void* const* d_in, const int* in_sizes, int n_in,
                              void* d_out, int out_size, void* d_ws, size_t ws_size,
                              hipStream_t stream) {
  const float* x     = (const float*)d_in[0];
  const float* w     = (const float*)d_in[1];
  const float* bias  = (const float*)d_in[2];
  const float* alpha = (const float*)d_in[3];
  float* ws  = (float*)d_ws;
  float* out = (float*)d_out;

  const int smem_fft    = 7 * 4096 * (int)sizeof(float);   // 112KB
  const int smem_cayley = 49152 * (int)sizeof(float);      // 192KB (<= 320KB/WGP)
  const int smem_out    = 7 * 4096 * (int)sizeof(float);

  // opt in to >64KB dynamic LDS (idempotent; ignore errors)
  (void)hipFuncSetAttribute((const void*)k_fft,
      hipFuncAttributeMaxDynamicSharedMemorySize, smem_fft);
  (void)hipFuncSetAttribute((const void*)k_cayley,
      hipFuncAttributeMaxDynamicSharedMemorySize, smem_cayley);
  (void)hipFuncSetAttribute((const void*)k_out,
      hipFuncAttributeMaxDynamicSharedMemorySize, smem_out);

  (void)hipMemsetAsync(ws + OFF_NORM, 0, sizeof(float), stream);
  k_twiddle<<<16, 256, 0, stream>>>(ws);
  k_norm  <<<NFREQ, 256, 0, stream>>>(w, ws);
  k_fft   <<<BATCH * CIN, 256, smem_fft, stream>>>(x, ws);
  k_cayley<<<NFREQ, 256, smem_cayley, stream>>>(w, alpha, ws);
  k_out   <<<BATCH * COUT, 256, smem_out, stream>>>(bias, ws, out);
}